// ClipForegroundEstimator_24584392802327
// MI455X (gfx1250) — compile-verified
//
#include <hip/hip_runtime.h>
#include <hip/hip_bf16.h>

typedef float v2f __attribute__((ext_vector_type(2)));
typedef float v8f __attribute__((ext_vector_type(8)));
typedef int   v4i __attribute__((ext_vector_type(4)));

#define NGROUPS 32
#define CH_PER_G 16
#define GN_EPS 1e-5f
#define APAD 34   // even -> 8B-aligned b64 LDS reads; i*34 mod 64 distinct, conflict-free
#define BPAD 36   // even + row byte-stride 144 (16B mult) for async b128; conflict-free

#if defined(__has_builtin)
#  if __has_builtin(__builtin_amdgcn_global_load_async_to_lds_b128)
#    define ASYNC_LDS 1
#  endif
#endif
#ifndef ASYNC_LDS
#  define ASYNC_LDS 0
#endif

__device__ __forceinline__ void async_copy_b128(const float* gsrc, float* ldst)
{
#if ASYNC_LDS
    __builtin_amdgcn_global_load_async_to_lds_b128(
        (__attribute__((address_space(1))) v4i*)(void*)gsrc,
        (__attribute__((address_space(3))) v4i*)(void*)ldst, 0, 0);
#else
    (void)gsrc; (void)ldst;
#endif
}

__device__ __forceinline__ void wait_async0()
{
#if defined(__AMDGCN__)
#  if defined(__has_builtin) && __has_builtin(__builtin_amdgcn_s_wait_asynccnt)
    __builtin_amdgcn_s_wait_asynccnt(0);
#  else
    asm volatile("s_wait_asynccnt 0" ::: "memory");
#  endif
#endif
}

// ---------------------------------------------------------------------------
// Fused WMMA GEMM with optional fused GroupNorm-on-load and mask-on-epilogue:
//   Y[b,m,t] = act( mask[t] * sum_k A[m,k]*gn(X(b,k,t)) + bias[m] )
//   A       : [Mtot x K] row-major; rows >= Mvalid read as zero
//   X       : xlayout==0 -> X[b][k][t];  xlayout==1 -> X[b][t][k]
//   gnstats/gngamma/gnbeta (nullable): per-K-channel affine applied while
//             staging B (k-dependent only); gnmask (nullable): applied in the
//             epilogue (n-dependent only, commutes with the K-reduction).
//   stats   : optional (sum, sumsq) per (b, m/16) of the *output*, via atomics
// 256 threads = 8 wave32, 2(M) x 4(N) wave grid, 32x64 register tile per wave
// (2x4 WMMA accs), 64x256 workgroup tile, double-buffered LDS, one barrier
// per 32-K chunk. xlayout==1 B staging uses global_load_async_to_lds_b128.
// ---------------------------------------------------------------------------
__global__ __launch_bounds__(256)
void gemm_wmma_kernel(const float* __restrict__ A, const float* __restrict__ bias,
                      const float* __restrict__ X, float* __restrict__ Y,
                      float* __restrict__ stats,
                      const float* __restrict__ gnstats,
                      const float* __restrict__ gngamma,
                      const float* __restrict__ gnbeta,
                      const float* __restrict__ gnmask,
                      int Mtot, int Mvalid, int K, int T,
                      int xlayout, int do_sigmoid)
{
    __shared__ float Alds[2 * 64 * APAD];
    __shared__ float Blds[2 * 256 * BPAD];

    const int tid  = threadIdx.x;
    const int lane = tid & 31;
    const int wave = tid >> 5;
    const int half = lane >> 4;
    const int lsub = lane & 15;
    const int wm   = wave >> 2;     // 0..1  (M wave)
    const int wn   = wave & 3;      // 0..3  (N wave)

    const int m0 = blockIdx.y * 64;
    const int t0 = blockIdx.x * 256;
    const int b  = blockIdx.z;

    const float inv_cnt = 1.f / (float)(CH_PER_G * T);

    v8f acc[2][4];
#pragma unroll
    for (int i = 0; i < 2; ++i)
#pragma unroll
        for (int j = 0; j < 4; ++j)
            acc[i][j] = v8f{0.f, 0.f, 0.f, 0.f, 0.f, 0.f, 0.f, 0.f};

    float4 pa[2], pb[8];

    auto loadA = [&](int k0) {
#pragma unroll
        for (int i = 0; i < 2; ++i) {
            const int q  = tid + (i << 8);
            const int m  = q >> 3;
            const int kq = (q & 7) << 2;
            const int gm = m0 + m;
            pa[i] = (gm < Mvalid)
                        ? *(const float4*)(A + (size_t)gm * K + k0 + kq)
                        : make_float4(0.f, 0.f, 0.f, 0.f);
        }
    };
    auto loadB0 = [&](int k0) {   // xlayout==0 global prefetch
#pragma unroll
        for (int i = 0; i < 8; ++i) {
            const int q    = tid + (i << 8);
            const int krow = q >> 6;
            const int nq   = (q & 63) << 2;
            pb[i] = *(const float4*)(X + ((size_t)b * K + (k0 + krow)) * T + t0 + nq);
        }
    };
    auto storeLds = [&](int buf, int k0c) {
        float* Ab = Alds + buf * (64 * APAD);
        float* Bb = Blds + buf * (256 * BPAD);
#pragma unroll
        for (int i = 0; i < 2; ++i) {
            const int q  = tid + (i << 8);
            const int m  = q >> 3;
            const int kq = (q & 7) << 2;
            float* d = Ab + m * APAD + kq;
            d[0] = pa[i].x; d[1] = pa[i].y; d[2] = pa[i].z; d[3] = pa[i].w;
        }
#pragma unroll
        for (int i = 0; i < 8; ++i) {
            const int q    = tid + (i << 8);
            const int krow = q >> 6;
            const int nq   = (q & 63) << 2;
            float sc = 1.f, sh = 0.f;
            if (gnstats != nullptr) {           // fused GroupNorm affine (k-dep)
                const int ch = k0c + krow;
                const int g  = ch >> 4;
                const float mu  = gnstats[(b * NGROUPS + g) * 2 + 0] * inv_cnt;
                const float ex2 = gnstats[(b * NGROUPS + g) * 2 + 1] * inv_cnt;
                sc = rsqrtf(ex2 - mu * mu + GN_EPS) * gngamma[ch];
                sh = gnbeta[ch] - mu * sc;
            }
            Bb[(nq + 0) * BPAD + krow] = pb[i].x * sc + sh;  // transpose to [n][k]
            Bb[(nq + 1) * BPAD + krow] = pb[i].y * sc + sh;
            Bb[(nq + 2) * BPAD + krow] = pb[i].z * sc + sh;
            Bb[(nq + 3) * BPAD + krow] = pb[i].w * sc + sh;
        }
    };
    auto storeLdsA = [&](int buf) {
        float* Ab = Alds + buf * (64 * APAD);
#pragma unroll
        for (int i = 0; i < 2; ++i) {
            const int q  = tid + (i << 8);
            const int m  = q >> 3;
            const int kq = (q & 7) << 2;
            float* d = Ab + m * APAD + kq;
            d[0] = pa[i].x; d[1] = pa[i].y; d[2] = pa[i].z; d[3] = pa[i].w;
        }
    };
    auto asyncB1 = [&](int k0, int buf) {   // xlayout==1: direct [n][k] DMA copy
        float* Bb = Blds + buf * (256 * BPAD);
#pragma unroll
        for (int i = 0; i < 8; ++i) {
            const int q  = tid + (i << 8);
            const int n  = q >> 3;
            const int kq = (q & 7) << 2;
            async_copy_b128(X + ((size_t)b * T + (t0 + n)) * K + k0 + kq,
                            Bb + n * BPAD + kq);
        }
    };
    auto loadB1 = [&](int k0) {   // register fallback for xlayout==1
#pragma unroll
        for (int i = 0; i < 8; ++i) {
            const int q  = tid + (i << 8);
            const int n  = q >> 3;
            const int kq = (q & 7) << 2;
            pb[i] = *(const float4*)(X + ((size_t)b * T + (t0 + n)) * K + k0 + kq);
        }
    };
    auto storeLdsB1 = [&](int buf) {
        float* Bb = Blds + buf * (256 * BPAD);
#pragma unroll
        for (int i = 0; i < 8; ++i) {
            const int q  = tid + (i << 8);
            const int n  = q >> 3;
            const int kq = (q & 7) << 2;
            float* d = Bb + n * BPAD + kq;
            d[0] = pb[i].x; d[1] = pb[i].y; d[2] = pb[i].z; d[3] = pb[i].w;
        }
    };
    auto compute = [&](int buf) {
        const float* Ab = Alds + buf * (64 * APAD);
        const float* Bb = Blds + buf * (256 * BPAD);
        const float* Ar0 = Ab + (wm * 32 + 0  + lsub) * APAD;
        const float* Ar1 = Ab + (wm * 32 + 16 + lsub) * APAD;
        const float* Br0 = Bb + (wn * 64 + 0  + lsub) * BPAD;
        const float* Br1 = Bb + (wn * 64 + 16 + lsub) * BPAD;
        const float* Br2 = Bb + (wn * 64 + 32 + lsub) * BPAD;
        const float* Br3 = Bb + (wn * 64 + 48 + lsub) * BPAD;
#pragma unroll
        for (int kk = 0; kk < 32; kk += 4) {
            const int kb = kk + 2 * half;
            v2f a0; a0.x = Ar0[kb]; a0.y = Ar0[kb + 1];
            v2f a1; a1.x = Ar1[kb]; a1.y = Ar1[kb + 1];
            v2f b0; b0.x = Br0[kb]; b0.y = Br0[kb + 1];
            v2f b1; b1.x = Br1[kb]; b1.y = Br1[kb + 1];
            v2f b2; b2.x = Br2[kb]; b2.y = Br2[kb + 1];
            v2f b3; b3.x = Br3[kb]; b3.y = Br3[kb + 1];
            acc[0][0] = __builtin_amdgcn_wmma_f32_16x16x4_f32(false, a0, false, b0, (short)0, acc[0][0], false, false);
            acc[0][1] = __builtin_amdgcn_wmma_f32_16x16x4_f32(false, a0, false, b1, (short)0, acc[0][1], false, false);
            acc[0][2] = __builtin_amdgcn_wmma_f32_16x16x4_f32(false, a0, false, b2, (short)0, acc[0][2], false, false);
            acc[0][3] = __builtin_amdgcn_wmma_f32_16x16x4_f32(false, a0, false, b3, (short)0, acc[0][3], false, false);
            acc[1][0] = __builtin_amdgcn_wmma_f32_16x16x4_f32(false, a1, false, b0, (short)0, acc[1][0], false, false);
            acc[1][1] = __builtin_amdgcn_wmma_f32_16x16x4_f32(false, a1, false, b1, (short)0, acc[1][1], false, false);
            acc[1][2] = __builtin_amdgcn_wmma_f32_16x16x4_f32(false, a1, false, b2, (short)0, acc[1][2], false, false);
            acc[1][3] = __builtin_amdgcn_wmma_f32_16x16x4_f32(false, a1, false, b3, (short)0, acc[1][3], false, false);
        }
    };

    const int nchunks = K >> 5;
    const bool useAsyncB = (xlayout == 1) && (ASYNC_LDS != 0);

    if (useAsyncB) {
        // B tile DMA'd straight to LDS (ASYNCcnt); A staged through 2 VGPR quads.
        loadA(0);
        asyncB1(0, 0);
        for (int c = 0; c < nchunks; ++c) {
            storeLdsA(c & 1);
            wait_async0();
            __syncthreads();
            if (c + 1 < nchunks) {
                loadA((c + 1) << 5);
                asyncB1((c + 1) << 5, (c + 1) & 1);
            }
            compute(c & 1);
        }
    } else if (xlayout == 0) {
        loadA(0);
        loadB0(0);
        for (int c = 0; c < nchunks; ++c) {
            storeLds(c & 1, c << 5);
            __syncthreads();
            if (c + 1 < nchunks) {
                loadA((c + 1) << 5);
                loadB0((c + 1) << 5);
            }
            compute(c & 1);
        }
    } else {
        loadA(0);
        loadB1(0);
        for (int c = 0; c < nchunks; ++c) {
            storeLdsA(c & 1);
            storeLdsB1(c & 1);
            __syncthreads();
            if (c + 1 < nchunks) {
                loadA((c + 1) << 5);
                loadB1((c + 1) << 5);
            }
            compute(c & 1);
        }
    }

    // ---- epilogue: mask (n-dep), bias, optional sigmoid, store, GN stats ----
#pragma unroll
    for (int i = 0; i < 2; ++i) {
        float s = 0.f, s2 = 0.f;
#pragma unroll
        for (int j = 0; j < 4; ++j) {
            const int gn = t0 + wn * 64 + j * 16 + lsub;
            const float mk = (gnmask != nullptr) ? gnmask[(size_t)b * T + gn] : 1.f;
#pragma unroll
            for (int r = 0; r < 8; ++r) {
                const int gm = m0 + wm * 32 + i * 16 + half * 8 + r;
                float y = acc[i][j][r] * mk;
                if (bias != nullptr && gm < Mvalid) y += bias[gm];
                if (do_sigmoid) y = 1.f / (1.f + __expf(-y));
                Y[((size_t)b * Mtot + gm) * T + gn] = y;
                s  += y;
                s2 += y * y;
            }
        }
        if (stats != nullptr) {
#pragma unroll
            for (int off = 16; off; off >>= 1) {
                s  += __shfl_xor(s,  off, 32);
                s2 += __shfl_xor(s2, off, 32);
            }
            if (lane == 0) {
                const int g = (m0 + wm * 32 + i * 16) >> 4;
                atomicAdd(&stats[(b * NGROUPS + g) * 2 + 0], s);
                atomicAdd(&stats[(b * NGROUPS + g) * 2 + 1], s2);
            }
        }
    }
}

__global__ void zero_kernel(float* __restrict__ p, int n)
{
    const int i = blockIdx.x * blockDim.x + threadIdx.x;
    if (i < n) p[i] = 0.f;
}

// ---------------------------------------------------------------------------
// Top-k mean per (head z, batch b, class c): bitonic sort 2048 logits
// descending in LDS, k = max(1, valid_len / r_act), output sum(top k)/k.
// Logit buffers are [B][64][T] (row stride 64 with junk rows >= C, unread).
// ---------------------------------------------------------------------------
__global__ __launch_bounds__(256)
void topk_kernel(const float* __restrict__ textbuf, const float* __restrict__ clsbuf,
                 const float* __restrict__ img_masks, const float* __restrict__ masks,
                 float* __restrict__ out, int B, int C, int T, int r_act)
{
    __shared__ float vals[2048];
    __shared__ float red[256];

    const int c = blockIdx.x, b = blockIdx.y, z = blockIdx.z;
    const int tid = threadIdx.x;

    const float* src = (z == 0) ? (textbuf + ((size_t)b * 64 + c) * T)
                                : (clsbuf  + ((size_t)b * 64 + c) * T);
    const float* msk = (z == 0) ? (img_masks + (size_t)b * T)
                                : (masks     + (size_t)b * T);

    float mpart = 0.f;
    for (int i = tid; i < T; i += 256) {
        vals[i] = src[i];
        mpart  += msk[i];
    }
    red[tid] = mpart;
    __syncthreads();
    for (int s = 128; s; s >>= 1) {
        if (tid < s) red[tid] += red[tid + s];
        __syncthreads();
    }
    const int len = (int)(red[0] + 0.5f);
    int k = len / r_act;
    if (k < 1) k = 1;

    for (int size = 2; size <= T; size <<= 1) {
        for (int stride = size >> 1; stride > 0; stride >>= 1) {
            __syncthreads();
            for (int t = tid; t < T / 2; t += 256) {
                const int  pos  = 2 * t - (t & (stride - 1));
                const bool desc = ((pos & size) == 0);
                const float va = vals[pos], vb = vals[pos + stride];
                const bool sw = desc ? (va < vb) : (va > vb);
                if (sw) { vals[pos] = vb; vals[pos + stride] = va; }
            }
        }
    }
    __syncthreads();

    float s = 0.f;
    for (int i = tid; i < k; i += 256) s += vals[i];
    red[tid] = s;
    __syncthreads();
    for (int st = 128; st; st >>= 1) {
        if (tid < st) red[tid] += red[tid + st];
        __syncthreads();
    }
    if (tid == 0)
        out[((size_t)z * B + b) * C + c] = red[0] / (float)k;
}

// ---------------------------------------------------------------------------
extern "C" void kernel_launch(void* const* d_in, const int* in_sizes, int n_in,
                              void* d_out, int out_size, void* d_ws, size_t ws_size,
                              hipStream_t stream)
{
    const int B = 16, Fd = 2048, T = 2048, O = 512, Timg = 2048, D = 512, C = 20;

    const float* input  = (const float*)d_in[0];
    const float* masks  = (const float*)d_in[1];
    const float* tproto = (const float*)d_in[2];   // [1, C, D] -> [C][D]
    const float* imgf   = (const float*)d_in[3];   // [B, Timg, D]
    const float* imgm   = (const float*)d_in[4];   // [B, Timg]
    const float* W1 = (const float*)d_in[5];  const float* b1 = (const float*)d_in[6];
    const float* g1 = (const float*)d_in[7];  const float* be1 = (const float*)d_in[8];
    const float* W2 = (const float*)d_in[9];  const float* b2 = (const float*)d_in[10];
    const float* g2 = (const float*)d_in[11]; const float* be2 = (const float*)d_in[12];
    const float* Wc = (const float*)d_in[13]; const float* bc = (const float*)d_in[14];
    float* out = (float*)d_out;

    float* ws = (float*)d_ws;
    size_t off = 0;
    float* X1   = ws + off; off += (size_t)B * O * T;      // raw layer-1 out, 64 MB
    float* X2   = ws + off; off += (size_t)B * O * T;      // raw layer-2 out, 64 MB
    float* clsb = ws + off; off += (size_t)B * 64 * T;     // 8 MB
    float* txtb = ws + off; off += (size_t)B * 64 * Timg;  // 8 MB
    float* st1  = ws + off; off += (size_t)B * NGROUPS * 2;
    float* st2  = ws + off; off += (size_t)B * NGROUPS * 2; // contiguous after st1

    const int nstat = B * NGROUPS * 4;
    zero_kernel<<<dim3((nstat + 255) / 256), dim3(256), 0, stream>>>(st1, nstat);

    const dim3 blk(256);

    // layer 1: X1 = W1 @ input + b1 ; produce stats1 (no GN fused on input)
    gemm_wmma_kernel<<<dim3(T / 256, O / 64, B), blk, 0, stream>>>(
        W1, b1, input, X1, st1,
        nullptr, nullptr, nullptr, nullptr,
        O, O, Fd, T, /*xlayout=*/0, /*sigmoid=*/0);

    // layer 2: X2 = W2 @ (gn1(X1)*mask) + b2 ; consume stats1, produce stats2
    gemm_wmma_kernel<<<dim3(T / 256, O / 64, B), blk, 0, stream>>>(
        W2, b2, X1, X2, st2,
        st1, g1, be1, masks,
        O, O, O, T, 0, 0);

    // cls head: sigmoid(Wc @ (gn2(X2)*mask) + bc) ; consume stats2
    gemm_wmma_kernel<<<dim3(T / 256, 1, B), blk, 0, stream>>>(
        Wc, bc, X2, clsb, nullptr,
        st2, g2, be2, masks,
        64, C, O, T, 0, /*sigmoid=*/1);

    // text head: text_proto @ img_feats^T (xlayout=1 -> async LDS staging)
    gemm_wmma_kernel<<<dim3(Timg / 256, 1, B), blk, 0, stream>>>(
        tproto, nullptr, imgf, txtb, nullptr,
        nullptr, nullptr, nullptr, nullptr,
        64, C, D, Timg, /*xlayout=*/1, 0);

    // top-k mean for both heads -> out[2][B][C]
    topk_kernel<<<dim3(C, B, 2), blk, 0, stream>>>(
        txtb, clsb, imgm, masks, out, B, C, T, /*r_act=*/8);
}